// SubgraphProjection_81372450390359
// MI455X (gfx1250) — compile-verified
//
#include <hip/hip_runtime.h>

// SpMM: out[50000,128] = P @ X, P in sorted-row COO (CSR-like), via
// V_WMMA_F32_16X16X4_F32. One wave = 16-row x 64-col output tile:
//   A (16x4)  = value-weighted row-selection from row_idx/values
//   B (4x16)  = gathered X columns (4 column tiles of 16 -> 64 cols)
//   C/D       = fp32 16x16 accumulators, chained across nnz chunks of 4.
// Wave-uniform binary search (readfirstlane -> SGPR loop bounds, EXEC all-1s
// at every WMMA). 2-stage software pipeline enforced with sched_barrier(0):
// meta(i+2) and B(i+1) gather clauses issue BEFORE chunk-i WMMAs, so the
// WMMA waits tolerate the younger prefetches and each gather has a full
// iteration to hide L2 latency. All hot-loop addressing is 32-bit byte
// offsets from scalar bases (every buffer < 2^31 bytes).

typedef float v2f __attribute__((ext_vector_type(2)));
typedef float v8f __attribute__((ext_vector_type(8)));

#define DIM 128
#define ROWS_PER_TILE 16
#define COL_GROUPS 2                          // waves per row tile (column split)
#define COL_TILES (DIM / (16 * COL_GROUPS))   // 4 WMMA column tiles per wave
#define WAVES_PER_BLOCK 8

__global__ __launch_bounds__(WAVES_PER_BLOCK * 32)
void SubgraphProjection_spmm_wmma(const float* __restrict__ X,
                                  const float* __restrict__ vals,
                                  const int*   __restrict__ rows,
                                  const int*   __restrict__ cols,
                                  float* __restrict__ out,
                                  int nnz, int numSeg) {
  const int lane = threadIdx.x & 31;
  const int w    = blockIdx.x * WAVES_PER_BLOCK + (threadIdx.x >> 5);
  const int numTiles = (numSeg + ROWS_PER_TILE - 1) / ROWS_PER_TILE;
  if (w >= numTiles * COL_GROUPS) return;   // wave-uniform exit

  const int tile    = w / COL_GROUPS;
  const int colBase = (w % COL_GROUPS) * (16 * COL_TILES);
  const int rbase   = tile * ROWS_PER_TILE;
  const int rend    = (rbase + ROWS_PER_TILE < numSeg) ? rbase + ROWS_PER_TILE : numSeg;

  // CSR-like boundaries from sorted row_idx (uniform; force into SGPRs).
  int lo = 0, hi = nnz;
  while (lo < hi) { int mid = (lo + hi) >> 1; if (rows[mid] < rbase) lo = mid + 1; else hi = mid; }
  int s = lo;
  hi = nnz;
  while (lo < hi) { int mid = (lo + hi) >> 1; if (rows[mid] < rend) lo = mid + 1; else hi = mid; }
  int e = lo;
  s = __builtin_amdgcn_readfirstlane(s);
  e = __builtin_amdgcn_readfirstlane(e);

  v8f acc[COL_TILES] = {};

  const int      half = lane >> 4;   // 0: lanes 0-15, 1: lanes 16-31
  const unsigned l15  = lane & 15;
  const int      klo  = half * 2;    // this lane owns K pair {klo, klo+1} for A and B
  const int      mrow = rbase + (int)l15;     // A-fragment row owned by this lane
  const char* __restrict__ Xb = (const char*)(X + colBase); // scalar base
  const char* __restrict__ rB = (const char*)rows;
  const char* __restrict__ vB = (const char*)vals;
  const char* __restrict__ cB = (const char*)cols;

  const int nfull = (e - s) >> 2;    // full chunks of 4
  const int qend  = s + (nfull << 2);

  const unsigned mb = (unsigned)(s + klo) * 4u; // this lane's meta byte offset, chunk 0
  const unsigned bl = l15 * 4u;                 // lane byte offset within a B row

  if (nfull > 0) {
    // ---- prologue: meta(0) -> A(0), issue meta(1) (clamped), issue B(0) ----
    const int   r0 = *(const int*)  (rB + mb);
    const int   r1 = *(const int*)  (rB + mb + 4);
    const float v0 = *(const float*)(vB + mb);
    const float v1 = *(const float*)(vB + mb + 4);
    const unsigned c0 = *(const unsigned*)(cB + mb);
    const unsigned c1 = *(const unsigned*)(cB + mb + 4);
    v2f a;
    a.x = (r0 == mrow) ? v0 : 0.0f;
    a.y = (r1 == mrow) ? v1 : 0.0f;

    unsigned mo = (1 < nfull) ? mb + 16u : mb;  // clamped meta offset, chunk 1
    int      rn0 = *(const int*)  (rB + mo);
    int      rn1 = *(const int*)  (rB + mo + 4);
    float    vn0 = *(const float*)(vB + mo);
    float    vn1 = *(const float*)(vB + mo + 4);
    unsigned cn0 = *(const unsigned*)(cB + mo);
    unsigned cn1 = *(const unsigned*)(cB + mo + 4);

    float bx[COL_TILES], by[COL_TILES];
    {
      const unsigned o0 = (c0 << 9) + bl;   // c*DIM*4 bytes
      const unsigned o1 = (c1 << 9) + bl;
#pragma unroll
      for (int t = 0; t < COL_TILES; ++t) {
        bx[t] = *(const float*)(Xb + o0 + t * 64);
        by[t] = *(const float*)(Xb + o1 + t * 64);
      }
    }

    for (int i = 0; i < nfull; ++i) {
      // 1. next A from resident meta(i+1)
      v2f an;
      an.x = (rn0 == mrow) ? vn0 : 0.0f;
      an.y = (rn1 == mrow) ? vn1 : 0.0f;
      // 2. issue meta(i+2) loads (clamped; before B so waits stay staggered)
      const unsigned mo2 = (i + 2 < nfull) ? mb + 16u * (unsigned)(i + 2) : mb;
      const int      rt0 = *(const int*)  (rB + mo2);
      const int      rt1 = *(const int*)  (rB + mo2 + 4);
      const float    vt0 = *(const float*)(vB + mo2);
      const float    vt1 = *(const float*)(vB + mo2 + 4);
      const unsigned ct0 = *(const unsigned*)(cB + mo2);
      const unsigned ct1 = *(const unsigned*)(cB + mo2 + 4);
      // 3. issue B(i+1) gather clause from resident col indices
      float bxn[COL_TILES], byn[COL_TILES];
      {
        const unsigned o0 = (cn0 << 9) + bl;
        const unsigned o1 = (cn1 << 9) + bl;
#pragma unroll
        for (int t = 0; t < COL_TILES; ++t) {
          bxn[t] = *(const float*)(Xb + o0 + t * 64);
          byn[t] = *(const float*)(Xb + o1 + t * 64);
        }
      }
      // keep all prefetch issues ABOVE the WMMAs
      __builtin_amdgcn_sched_barrier(0);
      // 4. WMMAs for chunk i (wait on B(i) issued one iteration ago)
#pragma unroll
      for (int t = 0; t < COL_TILES; ++t) {
        v2f b; b.x = bx[t]; b.y = by[t];
        acc[t] = __builtin_amdgcn_wmma_f32_16x16x4_f32(
            false, a, false, b, (short)0, acc[t], false, false);
      }
      // rotate pipeline registers
      a = an;
#pragma unroll
      for (int t = 0; t < COL_TILES; ++t) { bx[t] = bxn[t]; by[t] = byn[t]; }
      rn0 = rt0; rn1 = rt1; vn0 = vt0; vn1 = vt1; cn0 = ct0; cn1 = ct1;
    }
  }

  // ---- tail chunk: 1-3 entries, branchless (clamped loads + selects) ----
  if (qend < e) {
    const int q0 = qend + klo, q1 = q0 + 1;
    const int qq0 = (q0 < e) ? q0 : e - 1;
    const int qq1 = (q1 < e) ? q1 : e - 1;
    const int      r0 = rows[qq0], r1 = rows[qq1];
    const float    v0 = vals[qq0], v1 = vals[qq1];
    const unsigned c0 = (unsigned)cols[qq0], c1 = (unsigned)cols[qq1];

    v2f a;
    a.x = ((q0 < e) && r0 == mrow) ? v0 : 0.0f;
    a.y = ((q1 < e) && r1 == mrow) ? v1 : 0.0f;

    const unsigned o0 = (c0 << 9) + bl;
    const unsigned o1 = (c1 << 9) + bl;
    float bx[COL_TILES], by[COL_TILES];
#pragma unroll
    for (int t = 0; t < COL_TILES; ++t) {
      bx[t] = *(const float*)(Xb + o0 + t * 64);
      by[t] = *(const float*)(Xb + o1 + t * 64);
    }
#pragma unroll
    for (int t = 0; t < COL_TILES; ++t) {
      v2f b; b.x = bx[t]; b.y = by[t];
      acc[t] = __builtin_amdgcn_wmma_f32_16x16x4_f32(
          false, a, false, b, (short)0, acc[t], false, false);
    }
  }

  // ---- store: C/D 16x16 f32 layout -> VGPR j = row (j + half*8), N = l15 ----
  const unsigned orow0 = (unsigned)(rbase + half * 8);
#pragma unroll
  for (int t = 0; t < COL_TILES; ++t) {
#pragma unroll
    for (int j = 0; j < 8; ++j) {
      const int row = rbase + j + half * 8;
      if (row < numSeg)
        out[(orow0 + j) * (unsigned)DIM + (unsigned)colBase + t * 16 + l15] = acc[t][j];
    }
  }
}

extern "C" void kernel_launch(void* const* d_in, const int* in_sizes, int n_in,
                              void* d_out, int out_size, void* d_ws, size_t ws_size,
                              hipStream_t stream) {
  // setup_inputs() order: input_matrix(f32), values(f32), row_idx(i32), col_idx(i32)
  const float* X    = (const float*)d_in[0];
  const float* vals = (const float*)d_in[1];
  const int*   rows = (const int*)d_in[2];
  const int*   cols = (const int*)d_in[3];
  float* out = (float*)d_out;

  const int nnz      = in_sizes[1];
  const int numSeg   = out_size / DIM;
  const int numTiles = (numSeg + ROWS_PER_TILE - 1) / ROWS_PER_TILE;
  const int totalWaves = numTiles * COL_GROUPS;
  const int blocks = (totalWaves + WAVES_PER_BLOCK - 1) / WAVES_PER_BLOCK;

  SubgraphProjection_spmm_wmma<<<blocks, WAVES_PER_BLOCK * 32, 0, stream>>>(
      X, vals, rows, cols, out, nnz, numSeg);
}